// Upsample_42142219108416
// MI455X (gfx1250) — compile-verified
//
#include <hip/hip_runtime.h>

// Upsample 2x with 4x4 separable FIR (upfirdn2d), fp32.
// Input  : (B*C, 256, 256) = 512 images
// Output : (B*C, 512, 512)
//
// Bandwidth-bound (>=640MB traffic, ~1 GFLOP -> HBM roofline ~28us at
// 23.3 TB/s). Strategy: async global->LDS tile staging (CDNA5
// GLOBAL_LOAD_ASYNC_TO_LDS_B32, ASYNCcnt + s_wait_asynccnt), 2x2 output
// blocks from 3x3 LDS neighborhoods, non-temporal float2 stores so the
// write-once 512MB output does not evict the reusable input from L2.

#define H_IN   256
#define W_IN   256
#define H_OUT  (2 * H_IN)
#define W_OUT  (2 * W_IN)

#define TILE_OH 64   // output rows per block
#define TILE_OW 64   // output cols per block
#define TILE_IH 32   // input core rows  (TILE_OH / 2)
#define TILE_IW 32   // input core cols  (TILE_OW / 2)
#define LDS_H   34   // TILE_IH + 2 halo rows
#define LDS_W   34   // TILE_IW + 2 halo cols
#define LDS_STRIDE 36 // padded to dodge bank conflicts

#define NTHREADS 256

typedef float f32x2 __attribute__((ext_vector_type(2)));

__global__ __launch_bounds__(NTHREADS) void upsample2x_fir_kernel(
    const float* __restrict__ in,
    const float* __restrict__ kern,   // 4x4 row-major, as passed by harness
    float* __restrict__ out)
{
    __shared__ float tile[LDS_H * LDS_STRIDE];

    const int tid   = threadIdx.x;
    const int tileX = blockIdx.x;     // along output width
    const int tileY = blockIdx.y;     // along output height
    const int img   = blockIdx.z;     // b*C + c

    const int inIY = tileY * TILE_IH; // input-core origin
    const int inIX = tileX * TILE_IW;

    const float* imgIn  = in  + (size_t)img * (H_IN * W_IN);
    float*       imgOut = out + (size_t)img * ((size_t)H_OUT * W_OUT);

    // ---- Stage (LDS_H x LDS_W) input tile (halo = clamped coords) into LDS
    //      via CDNA5 async global->LDS copies. Tracked by ASYNCcnt. ----
    for (int e = tid; e < LDS_H * LDS_W; e += NTHREADS) {
        const int ly = e / LDS_W;
        const int lx = e - ly * LDS_W;
        int gy = inIY - 1 + ly;  gy = min(max(gy, 0), H_IN - 1);
        int gx = inIX - 1 + lx;  gx = min(max(gx, 0), W_IN - 1);
        // LDS byte address (wave-relative): low 32 bits of the flat shared ptr
        const unsigned ldsAddr =
            (unsigned)(size_t)(&tile[ly * LDS_STRIDE + lx]);
        const unsigned long long ga =
            (unsigned long long)(size_t)(imgIn + gy * W_IN + gx);
        asm volatile("global_load_async_to_lds_b32 %0, %1, off"
                     :: "v"(ldsAddr), "v"(ga)
                     : "memory");
    }
    asm volatile("s_wait_asynccnt 0" ::: "memory");
    __syncthreads();

    // ---- 4x4 kernel weights (uniform; small cached loads) ----
    float k[16];
    #pragma unroll
    for (int i = 0; i < 16; ++i) k[i] = kern[i];

    // Thread -> 2x2 output blocks: qx in [0,32), qy in {qy0, qy0+8, +16, +24}
    const int qx  = tid & 31;
    const int qy0 = tid >> 5;

    #pragma unroll
    for (int h = 0; h < 4; ++h) {
        const int qy = qy0 + h * 8;   // 0..31

        // 3x3 input neighborhood (LDS rows qy..qy+2, cols qx..qx+2),
        // masked to honor the zero-padding of upfirdn2d at image borders.
        float v[3][3];
        #pragma unroll
        for (int dy = 0; dy < 3; ++dy) {
            const int gy = inIY + qy - 1 + dy;
            const bool rv = (gy >= 0) && (gy < H_IN);
            #pragma unroll
            for (int dx = 0; dx < 3; ++dx) {
                const int gx = inIX + qx - 1 + dx;
                const bool cv = (gx >= 0) && (gx < W_IN);
                const float t = tile[(qy + dy) * LDS_STRIDE + (qx + dx)];
                v[dy][dx] = (rv && cv) ? t : 0.0f;
            }
        }

        // Parity-phase taps (kernel applied flipped; indices pre-folded):
        //  even y -> K rows {3 (dy=0), 1 (dy=1)}; odd y -> {2 (dy=1), 0 (dy=2)}
        //  even x -> K cols {3 (dx=0), 1 (dx=1)}; odd x -> {2 (dx=1), 0 (dx=2)}
        const float o00 = k[3*4+3]*v[0][0] + k[3*4+1]*v[0][1]
                        + k[1*4+3]*v[1][0] + k[1*4+1]*v[1][1];
        const float o01 = k[3*4+2]*v[0][1] + k[3*4+0]*v[0][2]
                        + k[1*4+2]*v[1][1] + k[1*4+0]*v[1][2];
        const float o10 = k[2*4+3]*v[1][0] + k[2*4+1]*v[1][1]
                        + k[0*4+3]*v[2][0] + k[0*4+1]*v[2][1];
        const float o11 = k[2*4+2]*v[1][1] + k[2*4+0]*v[1][2]
                        + k[0*4+2]*v[2][1] + k[0*4+0]*v[2][2];

        const int oy = tileY * TILE_OH + 2 * qy;
        const int ox = tileX * TILE_OW + 2 * qx;

        // Write-once output: non-temporal stores (global_store_b64 th:NT)
        // keep the 512MB output stream from evicting the input in L2.
        f32x2 r0 = {o00, o01};
        f32x2 r1 = {o10, o11};
        __builtin_nontemporal_store(
            r0, reinterpret_cast<f32x2*>(imgOut + (size_t)oy * W_OUT + ox));
        __builtin_nontemporal_store(
            r1, reinterpret_cast<f32x2*>(imgOut + (size_t)(oy + 1) * W_OUT + ox));
    }
}

extern "C" void kernel_launch(void* const* d_in, const int* in_sizes, int n_in,
                              void* d_out, int out_size, void* d_ws, size_t ws_size,
                              hipStream_t stream) {
    const float* x    = (const float*)d_in[0];   // (4,128,256,256) fp32
    const float* kern = (const float*)d_in[1];   // (4,4) fp32
    float* out = (float*)d_out;                  // (4,128,512,512) fp32

    const int images = in_sizes[0] / (H_IN * W_IN); // 512
    dim3 grid(W_OUT / TILE_OW, H_OUT / TILE_OH, images); // (8,8,512)
    upsample2x_fir_kernel<<<grid, NTHREADS, 0, stream>>>(x, kern, out);
}